// PatchDivider_57621281243393
// MI455X (gfx1250) — compile-verified
//
#include <hip/hip_runtime.h>
#include <stdint.h>

// ---- problem constants (match reference) ----
#define B_      32
#define N_      131072
#define PATCH   64
#define L_      (N_ / PATCH)        // 2048
#define NPATCH  (B_ * L_)           // 65536 patches total

// ---- tiling ----
#define WAVES_PER_BLOCK    4
#define THREADS_PER_BLOCK  (WAVES_PER_BLOCK * 32)
#define PATCHES_PER_WAVE   16
#define PATCHES_PER_BLOCK  (WAVES_PER_BLOCK * PATCHES_PER_WAVE)   // 64
#define PTS_PER_WAVE       (PATCHES_PER_WAVE * PATCH)             // 1024

// LDS layout: one patch = 64 points * 3 floats = 192 dwords; pad to 196 dwords
// (784 B) so lane L's phase-2 read address (L*196 dw) maps to bank 4L -- the 16
// lanes of a half-wave land on 16 distinct banks.
#define PATCH_STRIDE_DW  196
#define LDS_DW_PER_WAVE  (PATCHES_PER_WAVE * PATCH_STRIDE_DW)     // 3136 dwords

typedef __attribute__((ext_vector_type(2))) float v2f;   // A/B frag: 16x4 / 4x16 f32 = 2 VGPRs
typedef __attribute__((ext_vector_type(8))) float v8f;   // C/D frag: 16x16 f32 = 8 VGPRs

__global__ __launch_bounds__(THREADS_PER_BLOCK)
void patch_divide_kernel(const float* __restrict__ pts,        // [B*N, 3]
                         const int*   __restrict__ orders,     // [B*N]
                         float*       __restrict__ outPatches, // [B,L,64,3] flat
                         float*       __restrict__ outCenters) // [B,L,3] flat
{
    __shared__ float lds[WAVES_PER_BLOCK * LDS_DW_PER_WAVE];

    const int tid  = threadIdx.x;
    const int wave = tid >> 5;
    const int lane = tid & 31;
    float* wlds = lds + wave * LDS_DW_PER_WAVE;   // wave-private tile -> no barriers

    const long long wavePatch0 = (long long)blockIdx.x * PATCHES_PER_BLOCK
                               + (long long)wave * PATCHES_PER_WAVE;
    const long long j0 = wavePatch0 * PATCH;      // first global point slot of this tile

    // ---------------- Phase 1: async gather of 1024 permuted points into LDS ----
    // Indices: int4 per lane -> 8 fully-coalesced b128 loads (512B/wave/instr).
    // Points:  GLOBAL_LOAD_ASYNC_TO_LDS_B64 + _B32 offset:8 per 12-byte point;
    //          per-lane global address, per-lane LDS dest, tracked by ASYNCcnt.
    //          No VGPR round-trip, no LOADcnt stalls; one s_wait_asynccnt 0 total.
    // The gather is random but pts (48 MiB) is resident in the 192 MB L2.
    {
        const int4* ord4 = (const int4*)(orders + j0);
        #pragma unroll 2
        for (int i = 0; i < 8; ++i) {
            const int4 o  = ord4[lane + 32 * i];
            const int  t0 = 4 * lane + 128 * i;        // 4 consecutive points, same patch
            const int  m  = t0 >> 6;                   // local patch 0..15
            const int  pos = t0 & 63;                  // point-in-patch (multiple of 4)
            // Low 32 bits of a generic pointer into LDS == wave-relative LDS byte addr.
            const unsigned ldsBase =
                (unsigned)(uintptr_t)(wlds + m * PATCH_STRIDE_DW + pos * 3);
            const int idx[4] = {o.x, o.y, o.z, o.w};
            #pragma unroll
            for (int k = 0; k < 4; ++k) {
                const unsigned long long ga =
                    (unsigned long long)(uintptr_t)(pts + (long long)idx[k] * 3);
                const unsigned la = ldsBase + 12u * (unsigned)k;
                asm volatile("global_load_async_to_lds_b64 %0, %1, off"
                             :: "v"(la), "v"(ga) : "memory");
                asm volatile("global_load_async_to_lds_b32 %0, %1, off offset:8"
                             :: "v"(la), "v"(ga) : "memory");
            }
        }
    }
    // Fence the async LDS writes against the compiler-generated ds_loads below.
    asm volatile("s_wait_asynccnt 0" ::: "memory");

    // ---------------- Phase 2: per-patch sums via V_WMMA_F32_16X16X4_F32 --------
    // A[m,k] layout (32-bit A 16x4): lanes 0-15 supply K=0,1 (vgpr0,1) for row m=lane;
    // lanes 16-31 supply K=2,3 for row m=lane-16.  With B == ones(4x16):
    //   D[m,n] = sum_k A[m,k]   (row sum, replicated across all columns n)
    // Chunk c covers points 4c..4c+3; accumulating over 16 chunks yields the
    // 64-point sum for all 16 patches at once, one component per accumulator.
    v8f cx = {}, cy = {}, cz = {};
    const v2f ones = {1.0f, 1.0f};
    const int m    = lane & 15;
    const int hsel = lane >> 4;                    // 0: K=0,1  1: K=2,3
    const float* abase = wlds + m * PATCH_STRIDE_DW;

    #pragma unroll
    for (int c = 0; c < 16; ++c) {
        const float* p0 = abase + (4 * c + 2 * hsel) * 3;   // 2 consecutive points (24B)
        const float x0 = p0[0], y0 = p0[1], z0 = p0[2];
        const float x1 = p0[3], y1 = p0[4], z1 = p0[5];
        const v2f ax = {x0, x1}, ay = {y0, y1}, az = {z0, z1};
        cx = __builtin_amdgcn_wmma_f32_16x16x4_f32(false, ax, false, ones, (short)0, cx, false, false);
        cy = __builtin_amdgcn_wmma_f32_16x16x4_f32(false, ay, false, ones, (short)0, cy, false, false);
        cz = __builtin_amdgcn_wmma_f32_16x16x4_f32(false, az, false, ones, (short)0, cz, false, false);
    }
    // D layout: vgpr r holds S[r] on lanes 0-15 and S[r+8] on lanes 16-31,
    // replicated across all 16 columns -> every lane owns the 8 centers of its half.

    const float inv = 1.0f / (float)PATCH;

    // ---------------- Phase 3a: write centers -----------------------------------
    // Lane 0 owns patches 0..7, lane 16 owns 8..15; fully static v8f indexing.
    if ((lane & 15) == 0) {
        float* c = outCenters + (wavePatch0 + (long long)hsel * 8) * 3;
        #pragma unroll
        for (int r = 0; r < 8; ++r) {
            c[r * 3 + 0] = cx[r] * inv;
            c[r * 3 + 1] = cy[r] * inv;
            c[r * 3 + 2] = cz[r] * inv;
        }
    }

    // ---------------- Phase 3b: centered-point writes ----------------------------
    // Half-wave hsel handles local patches [8*hsel .. 8*hsel+7]; the needed centers
    // are exactly this half's cx/cy/cz[r].  16 lanes write 192B-contiguous runs.
    {
        const int hl = lane & 15;
        float*       outp    = outPatches + (j0 + (long long)hsel * 512) * 3;
        const float* ldsHalf = wlds + hsel * 8 * PATCH_STRIDE_DW;
        #pragma unroll
        for (int r = 0; r < 8; ++r) {
            const float mx = cx[r] * inv, my = cy[r] * inv, mz = cz[r] * inv;
            const float* pb = ldsHalf + r * PATCH_STRIDE_DW;
            float*       ob = outp + r * PATCH * 3;
            #pragma unroll
            for (int i = 0; i < 4; ++i) {
                const int pos = hl + 16 * i;
                const float* s = pb + pos * 3;
                float*       o = ob + pos * 3;
                o[0] = s[0] - mx;
                o[1] = s[1] - my;
                o[2] = s[2] - mz;
            }
        }
    }
}

extern "C" void kernel_launch(void* const* d_in, const int* in_sizes, int n_in,
                              void* d_out, int out_size, void* d_ws, size_t ws_size,
                              hipStream_t stream) {
    (void)in_sizes; (void)n_in; (void)out_size; (void)d_ws; (void)ws_size;
    const float* pts    = (const float*)d_in[0];
    const int*   orders = (const int*)d_in[1];
    float* outPatches = (float*)d_out;                     // B*N*3 floats
    float* outCenters = outPatches + (size_t)B_ * N_ * 3;  // then B*L*3 floats
    dim3 grid(NPATCH / PATCHES_PER_BLOCK);                 // 1024 blocks
    patch_divide_kernel<<<grid, THREADS_PER_BLOCK, 0, stream>>>(pts, orders,
                                                                outPatches, outCenters);
}